// aggregator_actor_61856118997743
// MI455X (gfx1250) — compile-verified
//
#include <hip/hip_runtime.h>
#include <hip/hip_bf16.h>

typedef float v2f __attribute__((ext_vector_type(2)));
typedef float v8f __attribute__((ext_vector_type(8)));

#define NUM_ENT 10000
#define SEQ 10
#define KMAX 400          // largest K over all GEMM call sites (attn_wc half = 200; wc full K=400 split)

// ---------------------------------------------------------------- elementwise
__global__ void k_zero(float* __restrict__ p, long n) {
  long stride = (long)gridDim.x * blockDim.x;
  for (long t = (long)blockIdx.x * blockDim.x + threadIdx.x; t < n; t += stride)
    p[t] = 0.f;
}

__global__ void k_gather(float* __restrict__ out, const float* __restrict__ tab,
                         const int* __restrict__ idx, long rows, int C, int mod) {
  long n = rows * (long)C;
  long stride = (long)gridDim.x * blockDim.x;
  for (long t = (long)blockIdx.x * blockDim.x + threadIdx.x; t < n; t += stride) {
    long r = t / C; int c = (int)(t - r * (long)C);
    long j = idx ? (long)idx[r] : r;
    if (mod > 0) j %= mod;
    out[t] = tab[j * C + c];
  }
}

// acc[dst[e],c] += feat[src[e],c] - (sub ? sub[e,c] : 0) ; deg[dst[e]] += 1
__global__ void k_seg_accum(float* __restrict__ acc, float* __restrict__ deg,
                            const float* __restrict__ feat, const float* __restrict__ sub,
                            const int* __restrict__ src, const int* __restrict__ dst,
                            long E, int C) {
  long n = E * (long)C;
  long stride = (long)gridDim.x * blockDim.x;
  for (long t = (long)blockIdx.x * blockDim.x + threadIdx.x; t < n; t += stride) {
    long e = t / C; int c = (int)(t - e * (long)C);
    int d = dst[e];
    float v = feat[(long)src[e] * C + c];
    if (sub) v -= sub[t];
    atomicAdd(&acc[(long)d * C + c], v);
    if (c == 0) atomicAdd(&deg[d], 1.f);
  }
}

__global__ void k_mean(float* __restrict__ acc, const float* __restrict__ deg,
                       long rows, int C) {
  long n = rows * (long)C;
  long stride = (long)gridDim.x * blockDim.x;
  for (long t = (long)blockIdx.x * blockDim.x + threadIdx.x; t < n; t += stride) {
    long r = t / C;
    acc[t] /= fmaxf(deg[r], 1.f);
  }
}

// ---------------------------------------------------------------- WMMA GEMM
// D[M,N] = act( A[M,K] @ B[K,N] + bias? + addend? ), fp32 via V_WMMA_F32_16X16X4_F32.
// Block = 256 threads = 8 waves; each block owns 8 consecutive M-tiles x 1 N-tile.
// The K x 16 B-panel is staged once into LDS in WMMA *fragment order* via
// CDNA5 async global->LDS copies (ASYNCcnt), so the K-loop is just:
//   global_load_b64 (A frag) + ds_load_b64 (B frag) + v_wmma_f32_16x16x4_f32
// with EXEC fully converged throughout.
// Requirements: M % 16 == 0 (true at all call sites), K % 4 == 0, N ragged OK.
// act: 0 none, 1 relu, 2 tanh
__global__ void __launch_bounds__(256)
k_gemm_f32_wmma(const float* __restrict__ A, const float* __restrict__ B,
                const float* __restrict__ bias, const float* __restrict__ addend,
                float* __restrict__ D, int M, int N, int K, int act) {
  // fragment-ordered B panel: slot = k4*64 + lane*2 + j  <->  B[4*k4 + (lane>>4)*2 + j, n0 + (lane&15)]
  __shared__ float sB[KMAX * 16];

  const int tid    = threadIdx.x;
  const int tilesN = (N + 15) >> 4;
  const int gM     = blockIdx.x / tilesN;      // group of 8 M-tiles
  const int tN     = blockIdx.x - gM * tilesN;
  const int n0     = tN * 16;

  // ---- stage B panel (async copy global -> LDS; zero-fill ragged columns)
  const int nslots = (K >> 2) * 64;            // == K*16
  for (int s = tid; s < nslots; s += 256) {
    int k4   = s >> 6;
    int rem  = s & 63;
    int ln   = rem >> 1;
    int j    = rem & 1;
    int k    = k4 * 4 + ((ln >> 4) << 1) + j;
    int n    = n0 + (ln & 15);
    if (n < N) {
      unsigned ldsoff = (unsigned)(size_t)(&sB[s]);       // LDS offset lives in low 32 bits
      const float* gp = B + (long)k * N + n;
      asm volatile("global_load_async_to_lds_b32 %0, %1, off"
                   :: "v"(ldsoff), "v"(gp) : "memory");
    } else {
      sB[s] = 0.f;
    }
  }
  asm volatile("s_wait_asynccnt 0x0" ::: "memory");
  __syncthreads();

  // ---- per-wave 16x16 tile
  const int wave = tid >> 5;
  const int lane = tid & 31;
  const int m0   = (gM * 8 + wave) * 16;
  if (m0 >= M) return;                         // wave-uniform; EXEC full below
  const int  half = lane >> 4;
  const int  l16  = lane & 15;
  const long rowA = (long)(m0 + l16) * K;
  const int  bOff = lane * 2;

  v8f acc = {};
  for (int k4 = 0; k4 < (K >> 2); ++k4) {
    const int ka = (k4 << 2) + (half << 1);
    v2f a, b;
    a[0] = A[rowA + ka];
    a[1] = A[rowA + ka + 1];
    __builtin_prefetch(&A[rowA + ka + 16], 0, 1);
    b[0] = sB[(k4 << 6) + bOff];
    b[1] = sB[(k4 << 6) + bOff + 1];
    acc = __builtin_amdgcn_wmma_f32_16x16x4_f32(false, a, false, b,
                                                (short)0, acc, false, false);
  }

  const int nB = n0 + l16;
  if (nB < N) {
    float bi = bias ? bias[nB] : 0.f;
#pragma unroll
    for (int v = 0; v < 8; ++v) {
      int m = m0 + v + half * 8;               // C/D layout: VGPR v = row v (+8 for hi half)
      long off = (long)m * N + nB;
      float x = acc[v] + bi;
      if (addend) x += addend[off];
      if (act == 1) x = fmaxf(x, 0.f);
      else if (act == 2) x = tanhf(x);
      D[off] = x;
    }
  }
}

// ---------------------------------------------------------------- attention
// one wave32 per query row m: scores_k = mask * <QA[m], wh[idx_k]>, softmax, ctx.
__global__ void k_attn(const float* __restrict__ QA, const float* __restrict__ wh,
                       const int* __restrict__ eidx, const int* __restrict__ emask,
                       const int* __restrict__ ridx, const int* __restrict__ rmask,
                       float* __restrict__ ctx, int Mtot, int Nent, int MQ, int Hdim) {
  int m = blockIdx.x * (blockDim.x >> 5) + (threadIdx.x >> 5);
  if (m >= Mtot) return;
  int lane = threadIdx.x & 31;
  const int *ip, *mp;
  if (m < Nent) { ip = eidx + (long)m * MQ;          mp = emask + (long)m * MQ; }
  else          { ip = ridx + (long)(m - Nent) * MQ; mp = rmask + (long)(m - Nent) * MQ; }
  const float* qa = QA + (long)m * Hdim;
  float sc[8];
  for (int k = 0; k < MQ; ++k) {
    const float* w = wh + (long)ip[k] * Hdim;
    float s = 0.f;
    for (int c = lane; c < Hdim; c += 32) s += qa[c] * w[c];
    for (int off = 16; off; off >>= 1) s += __shfl_down(s, off, 32);
    s = __shfl(s, 0, 32);
    sc[k] = s * (float)mp[k];  // masked features => masked score (exp(0) stays in softmax)
  }
  float mx = sc[0];
  for (int k = 1; k < MQ; ++k) mx = fmaxf(mx, sc[k]);
  float den = 0.f;
  for (int k = 0; k < MQ; ++k) { sc[k] = expf(sc[k] - mx); den += sc[k]; }
  float inv = 1.f / den;
  for (int c = lane; c < Hdim; c += 32) {
    float a = 0.f;
    for (int k = 0; k < MQ; ++k)
      a += sc[k] * (float)mp[k] * wh[(long)ip[k] * Hdim + c];
    ctx[(long)m * Hdim + c] = a * inv;
  }
}

// ---------------------------------------------------------------- scatters
__global__ void k_scatter_node(const float* __restrict__ h, const int* __restrict__ slot,
                               float* __restrict__ out, long N, int C, int blk) {
  long n = N * (long)C;
  long stride = (long)gridDim.x * blockDim.x;
  for (long t = (long)blockIdx.x * blockDim.x + threadIdx.x; t < n; t += stride) {
    long r = t / C; int c = (int)(t - r * (long)C);
    int s = slot[r];
    int nid = s % NUM_ENT;
    int g   = s / blk;
    out[((long)nid * SEQ + g) * C + c] = h[t];
  }
}

__global__ void k_scatter_edge(const float* __restrict__ rr, const int* __restrict__ slot,
                               float* __restrict__ out, long R, int C, int per) {
  long n = R * (long)C;
  long stride = (long)gridDim.x * blockDim.x;
  for (long t = (long)blockIdx.x * blockDim.x + threadIdx.x; t < n; t += stride) {
    long r = t / C; int c = (int)(t - r * (long)C);
    int s = slot[r];
    int row = s / per, pos = s - row * per;
    out[((long)row * SEQ + pos) * C + c] = rr[t];
  }
}

// ---------------------------------------------------------------- host side
static inline unsigned ew_blocks(long n) {
  long b = (n + 255) / 256;
  if (b < 1) b = 1;
  if (b > 1048576) b = 1048576;
  return (unsigned)b;
}

static inline void launch_gemm(const float* A, const float* B, const float* bias,
                               const float* add, float* D, int M, int N, int K,
                               int act, hipStream_t s) {
  long tilesM = (M + 15) / 16, tilesN = (N + 15) / 16;
  long groupsM = (tilesM + 7) / 8;             // 8 M-tiles (8 waves) per block
  unsigned blocks = (unsigned)(groupsM * tilesN);
  k_gemm_f32_wmma<<<blocks, 256, 0, s>>>(A, B, bias, add, D, M, N, K, act);
}

extern "C" void kernel_launch(void* const* d_in, const int* in_sizes, int n_in,
                              void* d_out, int out_size, void* d_ws, size_t ws_size,
                              hipStream_t stream) {
  const float* ent_embeds  = (const float*)d_in[0];
  const float* rel_embeds  = (const float*)d_in[1];
  const float* word_embeds = (const float*)d_in[2];
  const float* gcn_w1      = (const float*)d_in[3];
  const float* gcn_b1      = (const float*)d_in[4];
  const float* gcn_w2      = (const float*)d_in[5];
  const float* gcn_b2      = (const float*)d_in[6];
  const float* cg1_wn      = (const float*)d_in[7];
  const float* cg1_wl      = (const float*)d_in[8];
  const float* cg1_wr      = (const float*)d_in[9];
  const float* cg2_wn      = (const float*)d_in[10];
  const float* cg2_wl      = (const float*)d_in[11];
  const float* cg2_wr      = (const float*)d_in[12];
  const float* attn_wa     = (const float*)d_in[13];
  const float* attn_wc     = (const float*)d_in[14];
  const int* node_slot     = (const int*)d_in[15];
  const int* edge_src      = (const int*)d_in[16];
  const int* edge_dst      = (const int*)d_in[17];
  const int* edge_type     = (const int*)d_in[18];
  const int* word_id       = (const int*)d_in[19];
  const int* wsrc          = (const int*)d_in[20];
  const int* wdst          = (const int*)d_in[21];
  const int* ent_word_idx  = (const int*)d_in[22];
  const int* ent_word_mask = (const int*)d_in[23];
  const int* rel_word_idx  = (const int*)d_in[24];
  const int* rel_word_mask = (const int*)d_in[25];
  const int* r_ids_graph   = (const int*)d_in[26];
  const int* edge_slot     = (const int*)d_in[27];

  const int N    = in_sizes[15];          // 50000
  const int E    = in_sizes[16];          // 200000
  const int Nw   = in_sizes[19];          // 30000
  const int Ew   = in_sizes[20];          // 300000
  const int Rsel = in_sizes[26];          // 8192
  const int H    = 200, OUT = 100, MQ = 8;
  const int Mall = N + Rsel;              // 58192
  const int NGl  = 2048;

  float* ws = (float*)d_ws;
  // arena (float offsets)
  float* bE   = ws +         0L;  // e0 then e2        (E*200 = 40,000,000)
  float* bH   = ws +  40000000L;  // h0 then h2        (N*200 = 10,000,000)
  float* bE1  = ws +  50000000L;  // e1                (E*100 = 20,000,000)
  float* bAGG = ws +  70000000L;  // seg-mean acc      (<= 10,000,000)
  float* bTMP = ws +  80000000L;  // gemm tmp / wg+wh1 (<= 10,000,000)
  float* bWH  = ws +  90000000L;  // word features wh  (Nw*200 = 6,000,000)
  float* bDEG = ws +  96000000L;  // degree            (<= 50,000)
  float* bH1  = ws +  96050000L;  // h1                (N*100 = 5,000,000)
  float* bQ   = ws + 101050000L;  // Q                 (Mall*200)
  float* bQA  = ws + 112700000L;  // QA
  float* bCTX = ws + 124350000L;  // ctx
  float* bOUT = ws + 136000000L;  // final attention output

  // ---------------- word GCN ----------------
  float* wg  = bTMP;               // (Nw, 100)
  float* wh1 = bTMP + 3000000L;    // (Nw, 200)
  k_gather<<<ew_blocks((long)Nw * OUT), 256, 0, stream>>>(wg, word_embeds, word_id, Nw, OUT, 0);
  k_zero<<<ew_blocks((long)Nw * OUT), 256, 0, stream>>>(bAGG, (long)Nw * OUT);
  k_zero<<<ew_blocks(Nw), 256, 0, stream>>>(bDEG, Nw);
  k_seg_accum<<<ew_blocks((long)Ew * OUT), 256, 0, stream>>>(bAGG, bDEG, wg, nullptr, wsrc, wdst, Ew, OUT);
  k_mean<<<ew_blocks((long)Nw * OUT), 256, 0, stream>>>(bAGG, bDEG, Nw, OUT);
  launch_gemm(bAGG, gcn_w1, gcn_b1, nullptr, wh1, Nw, H, OUT, 1, stream);

  k_zero<<<ew_blocks((long)Nw * H), 256, 0, stream>>>(bAGG, (long)Nw * H);
  k_zero<<<ew_blocks(Nw), 256, 0, stream>>>(bDEG, Nw);
  k_seg_accum<<<ew_blocks((long)Ew * H), 256, 0, stream>>>(bAGG, bDEG, wh1, nullptr, wsrc, wdst, Ew, H);
  k_mean<<<ew_blocks((long)Nw * H), 256, 0, stream>>>(bAGG, bDEG, Nw, H);
  launch_gemm(bAGG, gcn_w2, gcn_b2, nullptr, bWH, Nw, H, H, 1, stream);

  // ---------------- CompGCN layer 1 ----------------
  k_gather<<<ew_blocks((long)N * H), 256, 0, stream>>>(bH, ent_embeds, node_slot, N, H, NUM_ENT);
  k_gather<<<ew_blocks((long)E * H), 256, 0, stream>>>(bE, rel_embeds, edge_type, E, H, 0);
  k_zero<<<ew_blocks((long)N * H), 256, 0, stream>>>(bAGG, (long)N * H);
  k_zero<<<ew_blocks(N), 256, 0, stream>>>(bDEG, N);
  k_seg_accum<<<ew_blocks((long)E * H), 256, 0, stream>>>(bAGG, bDEG, bH, bE, edge_src, edge_dst, E, H);
  k_mean<<<ew_blocks((long)N * H), 256, 0, stream>>>(bAGG, bDEG, N, H);
  launch_gemm(bAGG, cg1_wn, nullptr, nullptr, bTMP, N, OUT, H, 0, stream);      // agg@wn
  launch_gemm(bH, cg1_wl, nullptr, bTMP, bH1, N, OUT, H, 1, stream);            // h1 = relu(h@wl + .)
  launch_gemm(bE, cg1_wr, nullptr, nullptr, bE1, E, OUT, H, 0, stream);         // e1 = e@wr

  // ---------------- CompGCN layer 2 ----------------
  k_zero<<<ew_blocks((long)N * OUT), 256, 0, stream>>>(bAGG, (long)N * OUT);
  k_zero<<<ew_blocks(N), 256, 0, stream>>>(bDEG, N);
  k_seg_accum<<<ew_blocks((long)E * OUT), 256, 0, stream>>>(bAGG, bDEG, bH1, bE1, edge_src, edge_dst, E, OUT);
  k_mean<<<ew_blocks((long)N * OUT), 256, 0, stream>>>(bAGG, bDEG, N, OUT);
  launch_gemm(bAGG, cg2_wn, nullptr, nullptr, bTMP, N, H, OUT, 0, stream);
  launch_gemm(bH1, cg2_wl, nullptr, bTMP, bH, N, H, OUT, 1, stream);            // h2 -> bH
  launch_gemm(bE1, cg2_wr, nullptr, nullptr, bE, E, H, OUT, 0, stream);         // e2 -> bE

  // ---------------- attention ----------------
  k_gather<<<ew_blocks((long)N * H), 256, 0, stream>>>(bQ, bH, nullptr, N, H, 0);             // Q[:N] = h2
  k_gather<<<ew_blocks((long)Rsel * H), 256, 0, stream>>>(bQ + (long)N * H, bE, r_ids_graph, Rsel, H, 0);
  launch_gemm(bQ, attn_wa, nullptr, nullptr, bQA, Mall, H, H, 0, stream);       // QA = Q @ wa
  {
    unsigned blocks = (unsigned)((Mall + 7) / 8);                               // 8 waves / block
    k_attn<<<blocks, 256, 0, stream>>>(bQA, bWH, ent_word_idx, ent_word_mask,
                                       rel_word_idx, rel_word_mask, bCTX, Mall, N, MQ, H);
  }
  // out = tanh([ctx|Q] @ attn_wc) = tanh(ctx @ Wc[0:200] + Q @ Wc[200:400])
  launch_gemm(bCTX, attn_wc, nullptr, nullptr, bOUT, Mall, H, H, 0, stream);
  launch_gemm(bQ, attn_wc + (long)H * H, nullptr, bOUT, bOUT, Mall, H, H, 2, stream);

  // ---------------- scatter outputs ----------------
  float* out_f = (float*)d_out;
  const long seq_sz = (long)NGl * SEQ * H;                  // 4,096,000 (embed_seq first)
  k_zero<<<ew_blocks(out_size), 256, 0, stream>>>(out_f, (long)out_size);
  k_scatter_edge<<<ew_blocks((long)Rsel * H), 256, 0, stream>>>(
      bOUT + (long)N * H, edge_slot, out_f, Rsel, H, Rsel / NGl);
  k_scatter_node<<<ew_blocks((long)N * H), 256, 0, stream>>>(
      bOUT, node_slot, out_f + seq_sz, N, H, N / SEQ);
}